// TAM_360777253619
// MI455X (gfx1250) — compile-verified
//
#include <hip/hip_runtime.h>
#include <math.h>

// ---------------- problem constants ----------------
#define T_SEG 8
#define C_CH  256
#define HH    56
#define WWD   56
#define HW    (HH * WWD)     // 3136
#define HW4   (HW / 4)       // 784 float4 per plane
#define NCLIP 8              // 64 / T
#define NC    (NCLIP * C_CH) // 2048 (n*c planes)
#define C4    64             // C/4
#define KK    3
#define EPSV  1e-5f

typedef float v8f __attribute__((ext_vector_type(8)));
typedef float v4f __attribute__((ext_vector_type(4)));
typedef float v2f __attribute__((ext_vector_type(2)));
typedef int   v4i __attribute__((ext_vector_type(4)));

// pointer casts matching the gfx1250 async-to-LDS builtin signature:
//   void (int4 addrspace(1)* src, int4 addrspace(3)* dst, imm off, imm cpol)
#define ASG_V4I(p) ((__attribute__((address_space(1))) v4i*)(p))
#define ASL_V4I(p) ((__attribute__((address_space(3))) v4i*)(p))

// =====================================================================
// Kernel 1: spatial mean pool  x[nt,c,h,w] -> pooled[(n*C+c)*T + t]
// One block per (nt,c) plane. On gfx1250 the plane is staged into LDS
// with GLOBAL_LOAD_ASYNC_TO_LDS_B128 (ASYNCcnt path, no VGPR staging),
// then reduced. Loads use default (RT) policy so x populates L2 for
// the fuse pass.
// =====================================================================
__global__ __launch_bounds__(256)
void tam_pool_kernel(const float* __restrict__ x, float* __restrict__ pooled) {
    const int p   = blockIdx.x;          // 0 .. 64*256-1   (ntIdx*C + c)
    const int tid = threadIdx.x;
    float s = 0.f;

#if __has_builtin(__builtin_amdgcn_global_load_async_to_lds_b128)
    __shared__ float tile[HW];           // 12.5 KB plane
    {
        const v4f* gsrc = (const v4f*)(x + (size_t)p * HW);
        v4f* ltile = (v4f*)tile;
#pragma unroll
        for (int r = 0; r < 3; ++r) {    // 3*256 = 768 of 784 float4
            const int i = r * 256 + tid;
            __builtin_amdgcn_global_load_async_to_lds_b128(
                ASG_V4I(&gsrc[i]), ASL_V4I(&ltile[i]), 0, 0);
        }
        if (tid < HW4 - 768) {           // 16-lane tail
            const int i = 768 + tid;
            __builtin_amdgcn_global_load_async_to_lds_b128(
                ASG_V4I(&gsrc[i]), ASL_V4I(&ltile[i]), 0, 0);
        }
#if __has_builtin(__builtin_amdgcn_s_wait_asynccnt)
        __builtin_amdgcn_s_wait_asynccnt(0);
#else
        asm volatile("s_wait_asynccnt 0" ::: "memory");
#endif
    }
    __syncthreads();
    for (int i = tid; i < HW4; i += 256) {
        v4f v = ((const v4f*)tile)[i];
        s += v.x + v.y + v.z + v.w;
    }
#else
    const v4f* xp = (const v4f*)(x + (size_t)p * HW);
    for (int i = tid; i < HW4; i += 256) {
        v4f v = xp[i];
        s += v.x + v.y + v.z + v.w;
    }
#endif

    __shared__ float red[256];
    red[tid] = s;
    __syncthreads();
    for (int off = 128; off > 0; off >>= 1) {
        if (tid < off) red[tid] += red[tid + off];
        __syncthreads();
    }
    if (tid == 0) {
        const int ntIdx = p / C_CH, c = p % C_CH;
        const int nn = ntIdx / T_SEG, t = ntIdx % T_SEG;
        pooled[(nn * C_CH + c) * T_SEG + t] = red[0] * (1.0f / (float)HW);
    }
}

// =====================================================================
// Kernel 2: G branch with V_WMMA_F32_16X16X4_F32 (wave32, one wave/block)
//   y    = relu(BN(pooled[16x8] @ g_w1^T[8x16]))        : 2 wmma k-steps
//   out2 = y[16x16] @ g_w2^T[16x16(pad from 3)]         : 4 wmma k-steps
//   kern = softmax(out2[:, 0:3])
// A layout (f32 16x4): lane<16 holds M=lane K={0,1}; lane>=16 M=lane-16 K={2,3}
// B layout (f32 4x16): lane<16 holds N=lane K={0,1}; lane>=16 N=lane-16 K={2,3}
// C/D layout: vgpr r: lane<16 -> (M=r, N=lane); lane>=16 -> (M=8+r, N=lane-16)
// =====================================================================
__global__ __launch_bounds__(32)
void tam_gbranch_kernel(const float* __restrict__ pooled,
                        const float* __restrict__ g_w1,
                        const float* __restrict__ gg, const float* __restrict__ gb,
                        const float* __restrict__ gm, const float* __restrict__ gv,
                        const float* __restrict__ g_w2,
                        float* __restrict__ kern) {
    const int lane = threadIdx.x;
    const int half = lane >> 4;       // 0: low half, 1: high half
    const int l    = lane & 15;
    const int r0   = blockIdx.x * 16; // row tile over the 2048 (n*c) rows

    // ---- GEMM1: A = pooled rows [r0..r0+15] x K=8, B = g_w1^T (8x16) ----
    const float* arow = pooled + (size_t)(r0 + l) * T_SEG;
    v2f a0, a1, b0, b1;
    a0[0] = arow[half * 2 + 0];           a0[1] = arow[half * 2 + 1];
    a1[0] = arow[4 + half * 2 + 0];       a1[1] = arow[4 + half * 2 + 1];
    // B[k][j] = g_w1[j*8 + k], column j = l lives in lane (j | half<<4)
    b0[0] = g_w1[l * 8 + half * 2 + 0];   b0[1] = g_w1[l * 8 + half * 2 + 1];
    b1[0] = g_w1[l * 8 + 4 + half * 2];   b1[1] = g_w1[l * 8 + 5 + half * 2];

    v8f acc = {};
    acc = __builtin_amdgcn_wmma_f32_16x16x4_f32(false, a0, false, b0,
                                                (short)0, acc, false, false);
    acc = __builtin_amdgcn_wmma_f32_16x16x4_f32(false, a1, false, b1,
                                                (short)0, acc, false, false);

    // BN + ReLU over output feature j = column = l (same for both lane halves)
    const float scale = gg[l] * rsqrtf(gv[l] + EPSV);
    const float shift = gb[l] - gm[l] * scale;
#pragma unroll
    for (int r = 0; r < 8; ++r) {
        float y = fmaf(acc[r], scale, shift);
        acc[r] = y > 0.f ? y : 0.f;
    }

    // ---- re-layout D -> A via LDS (y is 16x16) ----
    __shared__ float ylds[16 * 16];
#pragma unroll
    for (int r = 0; r < 8; ++r)
        ylds[(half * 8 + r) * 16 + l] = acc[r];
    __syncthreads();

    // ---- GEMM2: A = y (16x16), B = g_w2^T padded to 16 cols, K=16 ----
    v8f acc2 = {};
#pragma unroll
    for (int ks = 0; ks < 4; ++ks) {
        const int k0 = ks * 4 + half * 2;
        v2f a, b;
        a[0] = ylds[l * 16 + k0];
        a[1] = ylds[l * 16 + k0 + 1];
        // B[k][j] = (j<3) ? g_w2[j*16 + k] : 0
        b[0] = (l < KK) ? g_w2[l * 16 + k0]     : 0.f;
        b[1] = (l < KK) ? g_w2[l * 16 + k0 + 1] : 0.f;
        acc2 = __builtin_amdgcn_wmma_f32_16x16x4_f32(false, a, false, b,
                                                     (short)0, acc2, false, false);
    }

    __syncthreads();
#pragma unroll
    for (int r = 0; r < 8; ++r)
        ylds[(half * 8 + r) * 16 + l] = acc2[r];
    __syncthreads();

    // ---- per-row softmax over the 3 valid columns ----
    if (lane < 16) {
        const float v0 = ylds[lane * 16 + 0];
        const float v1 = ylds[lane * 16 + 1];
        const float v2 = ylds[lane * 16 + 2];
        const float m  = fmaxf(v0, fmaxf(v1, v2));
        const float e0 = expf(v0 - m), e1 = expf(v1 - m), e2 = expf(v2 - m);
        const float inv = 1.f / (e0 + e1 + e2);
        float* kp = kern + (size_t)(r0 + lane) * 3;
        kp[0] = e0 * inv; kp[1] = e1 * inv; kp[2] = e2 * inv;
    }
}

// =====================================================================
// Kernel 3: L branch conv1 (C -> C/4, K=3, pad 1) + BN + ReLU
// one thread per (n, o, t): 768-MAC dot; pooled is L2-resident (64 KB)
// =====================================================================
__global__ __launch_bounds__(256)
void tam_lconv1_kernel(const float* __restrict__ pooled,
                       const float* __restrict__ l_w1,
                       const float* __restrict__ lg, const float* __restrict__ lb,
                       const float* __restrict__ lm, const float* __restrict__ lv,
                       float* __restrict__ z) {
    const int id = blockIdx.x * 256 + threadIdx.x;   // n*64*8 = 4096
    if (id >= NCLIP * C4 * T_SEG) return;
    const int nn  = id >> 9;          // /512
    const int rem = id & 511;
    const int o   = rem >> 3;
    const int t   = rem & 7;
    const float* wbase = l_w1  + (size_t)o  * C_CH * KK;
    const float* pbase = pooled + (size_t)nn * C_CH * T_SEG;
    float s = 0.f;
    for (int c = 0; c < C_CH; ++c) {
        const float* w  = wbase + c * KK;
        const float* pp = pbase + c * T_SEG;
        if (t > 0) s = fmaf(w[0], pp[t - 1], s);
        s = fmaf(w[1], pp[t], s);
        if (t < T_SEG - 1) s = fmaf(w[2], pp[t + 1], s);
    }
    const float scale = lg[o] * rsqrtf(lv[o] + EPSV);
    const float y = fmaf(s - lm[o], scale, lb[o]);
    z[id] = y > 0.f ? y : 0.f;        // z[(nn*64+o)*8 + t]
}

// =====================================================================
// Kernel 4: L branch 1x1 conv (C/4 -> C) + sigmoid -> act[(n*C+c)*T + t]
// =====================================================================
__global__ __launch_bounds__(256)
void tam_lconv2_kernel(const float* __restrict__ z,
                       const float* __restrict__ l_w2,
                       float* __restrict__ act) {
    const int id = blockIdx.x * 256 + threadIdx.x;   // n*256*8 = 16384
    const int nn  = id >> 11;         // /2048
    const int rem = id & 2047;
    const int c   = rem >> 3;
    const int t   = rem & 7;
    const float* w  = l_w2 + (size_t)c * C4;
    const float* zb = z + (size_t)nn * C4 * T_SEG + t;
    float s = 0.f;
#pragma unroll 8
    for (int o = 0; o < C4; ++o) s = fmaf(w[o], zb[o * T_SEG], s);
    act[id] = 1.f / (1.f + expf(-s)); // act[(nn*256+c)*8 + t]
}

// =====================================================================
// Kernel 5 (the bandwidth kernel): gate + dynamic 3-tap temporal conv.
// Each thread owns one float4 column for all T=8 timesteps -> x read
// exactly once (128-bit loads), out written once.
// x loads are NON-TEMPORAL (last use: hit L2 lines left by the pooling
// pass, mark for eviction); out stores are NON-TEMPORAL so the 205 MB
// output stream does not evict x from the 192 MB L2 mid-pass.
// =====================================================================
__global__ __launch_bounds__(256)
void tam_fuse_kernel(const float* __restrict__ x,
                     const float* __restrict__ act,
                     const float* __restrict__ kern,
                     float* __restrict__ out) {
    const int p    = blockIdx.y;                       // 0..2047 = nn*256 + c
    const int pix4 = blockIdx.x * 256 + threadIdx.x;
    if (pix4 >= HW4) return;
    const int nn = p >> 8;
    const int c  = p & 255;

    const float* ap = act  + (size_t)p * T_SEG;        // uniform -> scalar loads
    const float* kp = kern + (size_t)p * 3;
    const float k0 = kp[0], k1 = kp[1], k2 = kp[2];

    const size_t tstride = (size_t)C_CH * HW4;         // float4 units per t step
    const size_t base = ((size_t)(nn * T_SEG) * C_CH + c) * HW4 + pix4;
    const v4f* x4 = (const v4f*)x;
    v4f*       o4 = (v4f*)out;

    // Prefetch all 8 strided time planes (gfx1250 global_prefetch_b8)
#pragma unroll
    for (int t = 0; t < T_SEG; ++t)
        __builtin_prefetch((const void*)&x4[base + (size_t)t * tstride], 0, 3);

    v4f g[T_SEG];
#pragma unroll
    for (int t = 0; t < T_SEG; ++t) {
        v4f v = __builtin_nontemporal_load(&x4[base + (size_t)t * tstride]);
        g[t] = v * ap[t];
    }
#pragma unroll
    for (int t = 0; t < T_SEG; ++t) {
        v4f r = k1 * g[t];
        if (t > 0)         r += k0 * g[t - 1];
        if (t < T_SEG - 1) r += k2 * g[t + 1];
        __builtin_nontemporal_store(r, &o4[base + (size_t)t * tstride]);
    }
}

// =====================================================================
extern "C" void kernel_launch(void* const* d_in, const int* in_sizes, int n_in,
                              void* d_out, int out_size, void* d_ws, size_t ws_size,
                              hipStream_t stream) {
    const float* x        = (const float*)d_in[0];
    const float* g_w1     = (const float*)d_in[1];
    const float* g_bn_g   = (const float*)d_in[2];
    const float* g_bn_b   = (const float*)d_in[3];
    const float* g_bn_m   = (const float*)d_in[4];
    const float* g_bn_v   = (const float*)d_in[5];
    const float* g_w2     = (const float*)d_in[6];
    const float* l_w1     = (const float*)d_in[7];
    const float* l_bn_g   = (const float*)d_in[8];
    const float* l_bn_b   = (const float*)d_in[9];
    const float* l_bn_m   = (const float*)d_in[10];
    const float* l_bn_v   = (const float*)d_in[11];
    const float* l_w2     = (const float*)d_in[12];
    float* out = (float*)d_out;

    // workspace carve-up (fp32): pooled | kern | act | z  (~168 KB total)
    float* pooled = (float*)d_ws;
    float* kern   = pooled + (size_t)NC * T_SEG;     // 16384
    float* act    = kern   + (size_t)NC * 3;         // + 6144
    float* z      = act    + (size_t)NC * T_SEG;     // + 16384

    tam_pool_kernel<<<NCLIP * T_SEG * C_CH, 256, 0, stream>>>(x, pooled);

    tam_gbranch_kernel<<<NC / 16, 32, 0, stream>>>(pooled, g_w1, g_bn_g, g_bn_b,
                                                   g_bn_m, g_bn_v, g_w2, kern);

    tam_lconv1_kernel<<<(NCLIP * C4 * T_SEG + 255) / 256, 256, 0, stream>>>(
        pooled, l_w1, l_bn_g, l_bn_b, l_bn_m, l_bn_v, z);

    tam_lconv2_kernel<<<(NC * T_SEG) / 256, 256, 0, stream>>>(z, l_w2, act);

    dim3 fgrid((HW4 + 255) / 256, NC);
    tam_fuse_kernel<<<fgrid, 256, 0, stream>>>(x, act, kern, out);
}